// Distance_48395691492053
// MI455X (gfx1250) — compile-verified
//
#include <hip/hip_runtime.h>
#include <hip/hip_bf16.h>

typedef __attribute__((ext_vector_type(2))) float v2f;
typedef __attribute__((ext_vector_type(8))) float v8f;

#define KNBR   32          // MAX_NEIGHBORS
#define NW     4           // waves per block
#define TILE   16
#define BIGF   1e30f
#define CUT2   25.0f       // CUTOFF_UPPER^2
#define CUTLO  0.5f        // CUTOFF_LOWER

// One wave owns 16 center rows. batch is sorted, so candidate columns for these
// rows form a contiguous range found by binary search. For each 16-col tile in
// that range:
//   d2 tile = WMMA( A=(x,y,z,1), B=(-2x,-2y,-2z,sq_j) ) + sq_i
// then lanes 0-15 each maintain a 32-entry nearest list for their row in LDS.
__global__ __launch_bounds__(NW * 32)
void radius_topk_wmma_kernel(const float* __restrict__ pos,
                             const int*   __restrict__ batch,
                             int N,
                             int*   __restrict__ outSrc,   // edge_index[0] (neighbors)
                             int*   __restrict__ outDst,   // edge_index[1] (centers)
                             float* __restrict__ outW,     // edge_weight
                             float* __restrict__ outV,     // edge_vec [NK,3]
                             float* __restrict__ outM)     // mask (0/1)
{
    __shared__ float tileD[NW][TILE][TILE];   // masked d2 tile, per-wave region
    __shared__ float lstD[NW][TILE][KNBR];    // candidate d2
    __shared__ int   lstI[NW][TILE][KNBR];    // candidate index

    const int lane = threadIdx.x & 31;
    const int wv   = threadIdx.x >> 5;
    const int m0   = (blockIdx.x * NW + wv) * TILE;
    if (m0 >= N) return;                       // wave-uniform exit

    const int  half = (lane >= 16) ? 1 : 0;
    const int  l16  = lane & 15;

    // ---- A fragment: row m0+l16, lane<16 holds (K0,K1)=(x,y), lane>=16 holds (K2,K3)=(z,1)
    const int arow = m0 + l16;
    const float ax = pos[arow * 3 + 0];
    const float ay = pos[arow * 3 + 1];
    const float az = pos[arow * 3 + 2];
    v2f afrag;
    afrag[0] = half ? az   : ax;
    afrag[1] = half ? 1.0f : ay;

    // ---- per-accumulator-VGPR row metadata (VGPR v -> row rbase+v)
    float sqr[8];
    int   br[8];
    const int rbase = m0 + (half ? 8 : 0);
#pragma unroll
    for (int v = 0; v < 8; ++v) {
        const int r = rbase + v;
        const float x = pos[r * 3 + 0], y = pos[r * 3 + 1], z = pos[r * 3 + 2];
        sqr[v] = x * x + y * y + z * z;
        br[v]  = batch[r];
    }

    // ---- batch is sorted: binary-search the contiguous candidate column range
    // [jLo, jHi) covering batches batch[m0] .. batch[m0+15].
    const int bFirst = batch[m0];
    const int bLast  = batch[m0 + TILE - 1];
    int lo = 0, hi = N;
    while (lo < hi) {                      // lower_bound(bFirst)
        const int mid = (lo + hi) >> 1;
        if (batch[mid] < bFirst) lo = mid + 1; else hi = mid;
    }
    const int jLo = lo;
    lo = 0; hi = N;
    while (lo < hi) {                      // upper_bound(bLast)
        const int mid = (lo + hi) >> 1;
        if (batch[mid] <= bLast) lo = mid + 1; else hi = mid;
    }
    const int jHi = lo;                    // exclusive
    const int t0  = jLo / TILE;
    const int t1  = (jHi + TILE - 1) / TILE;

    // ---- init per-row candidate lists (lanes 0-15 own rows)
    int   cnt = 0, worstP = 0;
    float worstV = -BIGF;
    if (lane < TILE) {
        for (int t = 0; t < KNBR; ++t) {
            lstD[wv][lane][t] = BIGF;
            lstI[wv][lane][t] = -1;
        }
    }

    for (int jt = t0; jt < t1; ++jt) {
        const int j0  = jt * TILE;
        const int col = j0 + l16;

        // prefetch next column tile (global_prefetch_b8)
        if (jt + 1 < t1) __builtin_prefetch(&pos[(col + TILE) * 3], 0, 0);

        const float bx = pos[col * 3 + 0];
        const float by = pos[col * 3 + 1];
        const float bz = pos[col * 3 + 2];
        const float sqc = bx * bx + by * by + bz * bz;

        // B fragment: col j0+l16, lane<16 holds (K0,K1)=(-2x,-2y), lane>=16 holds (K2,K3)=(-2z,sq_j)
        v2f bfrag;
        bfrag[0] = half ? (-2.0f * bz) : (-2.0f * bx);
        bfrag[1] = half ? sqc          : (-2.0f * by);

        const int bcol = batch[col];

        // d2 tile (minus sq_i): one matrix op per 16x16 tile, EXEC fully converged here
        v8f acc = {};
        acc = __builtin_amdgcn_wmma_f32_16x16x4_f32(
            /*neg_a=*/false, afrag, /*neg_b=*/false, bfrag,
            /*c_mod=*/(short)0, acc, /*reuse_a=*/false, /*reuse_b=*/false);

        // mask + stage to per-wave LDS tile; C layout: VGPR v = row rbase+v, N = l16
#pragma unroll
        for (int v = 0; v < 8; ++v) {
            const int  r     = rbase + v;
            const float d2   = acc[v] + sqr[v];
            const bool valid = (br[v] == bcol) & (r != col) & (d2 <= CUT2);
            tileD[wv][v + (half ? 8 : 0)][l16] = valid ? d2 : BIGF;
        }

        // lanes 0-15: scan 16 candidates for own row, maintain nearest-32 list
        if (lane < TILE) {
            for (int c = 0; c < TILE; ++c) {
                const float d = tileD[wv][lane][c];
                if (d < 0.5f * BIGF) {
                    const int j = j0 + c;
                    if (cnt < KNBR) {
                        lstD[wv][lane][cnt] = d;
                        lstI[wv][lane][cnt] = j;
                        if (d > worstV) { worstV = d; worstP = cnt; }
                        ++cnt;
                    } else if (d < worstV) {
                        lstD[wv][lane][worstP] = d;
                        lstI[wv][lane][worstP] = j;
                        worstV = -BIGF;
                        for (int t = 0; t < KNBR; ++t) {
                            const float dt = lstD[wv][lane][t];
                            if (dt > worstV) { worstV = dt; worstP = t; }
                        }
                    }
                }
            }
        }
    }

    // ---- sort (ascending d2, tie on lower index) and emit edges
    if (lane < TILE) {
        const int row = m0 + lane;
        for (int k = 0; k < KNBR; ++k) {
            float bd = lstD[wv][lane][k];
            int   bi = lstI[wv][lane][k];
            int   bp = k;
            for (int t = k + 1; t < KNBR; ++t) {
                const float dt = lstD[wv][lane][t];
                const int   it = lstI[wv][lane][t];
                if (dt < bd || (dt == bd && it < bi)) { bd = dt; bi = it; bp = t; }
            }
            lstD[wv][lane][bp] = lstD[wv][lane][k];
            lstI[wv][lane][bp] = lstI[wv][lane][k];
            lstD[wv][lane][k]  = bd;
            lstI[wv][lane][k]  = bi;

            const long long e = (long long)row * KNBR + k;
            float vx = 0.f, vy = 0.f, vz = 0.f, w = 0.f;
            bool  m  = false;
            if (bd < 0.5f * BIGF && bi >= 0) {
                vx = pos[bi * 3 + 0] - pos[row * 3 + 0];
                vy = pos[bi * 3 + 1] - pos[row * 3 + 1];
                vz = pos[bi * 3 + 2] - pos[row * 3 + 2];
                const float ss = vx * vx + vy * vy + vz * vz;
                w = (ss > 0.f) ? sqrtf(ss) : 0.f;
                m = (w >= CUTLO);
            }
            outSrc[e]       = m ? bi  : -1;
            outDst[e]       = m ? row : -1;
            outW[e]         = m ? w   : 0.f;
            outV[3 * e + 0] = m ? vx  : 0.f;
            outV[3 * e + 1] = m ? vy  : 0.f;
            outV[3 * e + 2] = m ? vz  : 0.f;
            outM[e]         = m ? 1.f : 0.f;
        }
    }
}

extern "C" void kernel_launch(void* const* d_in, const int* in_sizes, int n_in,
                              void* d_out, int out_size, void* d_ws, size_t ws_size,
                              hipStream_t stream) {
    const float* pos   = (const float*)d_in[0];  // [N,3] f32
    const int*   batch = (const int*)d_in[1];    // [N]   i32
    const int N = in_sizes[1];
    const long long NK = (long long)N * KNBR;

    // d_out layout (4-byte elements, concatenated in reference return order):
    // [0,NK) edge_index[0], [NK,2NK) edge_index[1], [2NK,3NK) weight,
    // [3NK,6NK) edge_vec, [6NK,7NK) mask
    int*   outSrc = (int*)d_out;
    int*   outDst = outSrc + NK;
    float* outW   = (float*)d_out + 2 * NK;
    float* outV   = (float*)d_out + 3 * NK;
    float* outM   = (float*)d_out + 6 * NK;

    const int rowsPerBlock = NW * TILE;
    const int grid = (N + rowsPerBlock - 1) / rowsPerBlock;
    radius_topk_wmma_kernel<<<grid, NW * 32, 0, stream>>>(
        pos, batch, N, outSrc, outDst, outW, outV, outM);
}